// calculate_moment_59768764891231
// MI455X (gfx1250) — compile-verified
//
#include <hip/hip_runtime.h>
#include <hip/hip_bf16.h>

typedef float v2f __attribute__((ext_vector_type(2)));
typedef float v4f __attribute__((ext_vector_type(4)));
typedef float v8f __attribute__((ext_vector_type(8)));

#define THREADS          256
#define VEC4_PER_BLOCK   32768           // 131072 floats per block = 2^17 elements
#define BLOCKS_PER_IMG   512             // 2^26 / 2^17
#define BLOCKS_PER_SLICE 64              // slice = one (image,b): 2^23 elements
// channel (H*W = 2^20 elements) = 8 blocks -> block j in slice: channel = j >> 3

__global__ __launch_bounds__(THREADS)
void moment_partials_kernel(const float* __restrict__ img0,
                            const float* __restrict__ img1,
                            float* __restrict__ partials) {
    const int gb  = blockIdx.x;                 // 0..1023
    const int img = gb >> 9;                    // 0 or 1
    const int blk = gb & 511;                   // block within image
    const float* src = (img == 0) ? img0 : img1;
    const v4f* __restrict__ p = (const v4f*)src + (size_t)blk * VEC4_PER_BLOCK;
    const int tid = threadIdx.x;

    v2f ones; ones.x = 1.0f; ones.y = 1.0f;
    v8f c1a = {}; v8f c1b = {};                 // accumulators for sum(x)
    v8f c2a = {}; v8f c2b = {};                 // accumulators for sum(x^2)

    #pragma unroll 4
    for (int i = 0; i < VEC4_PER_BLOCK / THREADS; ++i) {
        v4f x = __builtin_nontemporal_load(&p[tid + i * THREADS]);
        v2f a0; a0.x = x.x; a0.y = x.y;
        v2f a1; a1.x = x.z; a1.y = x.w;
        v2f q0 = a0 * a0;
        v2f q1 = a1 * a1;
        // D = A x ones + C : every C column accumulates the same row-sums,
        // so summing all of C at the end gives 16 * (sum of all A elements).
        c1a = __builtin_amdgcn_wmma_f32_16x16x4_f32(false, a0, false, ones,
                                                    (short)0, c1a, false, false);
        c1b = __builtin_amdgcn_wmma_f32_16x16x4_f32(false, a1, false, ones,
                                                    (short)0, c1b, false, false);
        c2a = __builtin_amdgcn_wmma_f32_16x16x4_f32(false, q0, false, ones,
                                                    (short)0, c2a, false, false);
        c2b = __builtin_amdgcn_wmma_f32_16x16x4_f32(false, q1, false, ones,
                                                    (short)0, c2b, false, false);
    }

    v8f c1 = c1a + c1b;
    v8f c2 = c2a + c2b;
    float p1 = ((c1[0] + c1[1]) + (c1[2] + c1[3])) +
               ((c1[4] + c1[5]) + (c1[6] + c1[7]));
    float p2 = ((c2[0] + c2[1]) + (c2[2] + c2[3])) +
               ((c2[4] + c2[5]) + (c2[6] + c2[7]));
    p1 *= 0.0625f;   // undo 16x column replication (exact, power of 2)
    p2 *= 0.0625f;

    __shared__ float s1[THREADS];
    __shared__ float s2[THREADS];
    s1[tid] = p1;
    s2[tid] = p2;
    __syncthreads();
    #pragma unroll
    for (int s = THREADS / 2; s > 0; s >>= 1) {
        if (tid < s) {
            s1[tid] += s1[tid + s];
            s2[tid] += s2[tid + s];
        }
        __syncthreads();
    }
    if (tid == 0) {
        partials[2 * gb + 0] = s1[0];
        partials[2 * gb + 1] = s2[0];
    }
}

__global__ __launch_bounds__(32)
void moment_final_kernel(const float* __restrict__ partials,
                         float* __restrict__ out) {
    __shared__ double mom[16];
    const int t = threadIdx.x;
    if (t < 16) {
        const int img = t >> 3;
        const int b   = t & 7;
        const double N     = 8388608.0;          // C*H*W = 2^23
        const double Nd    = 1048576.0;          // H*W   = 2^20
        const double denom = 70368744177664.0;   // (C*H*W)^2 = 2^46
        double S1 = 0.0, S2 = 0.0, S1d = 0.0, S2d = 0.0;
        const int base = img * BLOCKS_PER_IMG + b * BLOCKS_PER_SLICE;
        for (int j = 0; j < BLOCKS_PER_SLICE; ++j) {
            double a = (double)partials[2 * (base + j) + 0];
            double q = (double)partials[2 * (base + j) + 1];
            S1 += a; S2 += q;
            if ((j >> 3) == b) { S1d += a; S2d += q; }   // diagonal channel c==b
        }
        double mean = S1 / N;
        double var  = (S2 - N * mean * mean) / (N - 1.0);   // unbiased (ddof=1)
        double m    = (S2d - 2.0 * mean * S1d + Nd * mean * mean) / var / denom;
        mom[t] = m;
    }
    __syncthreads();
    if (t == 0) {
        double m1 = 0.0, m2 = 0.0;
        for (int i = 0; i < 8; ++i) { m1 += mom[i]; m2 += mom[8 + i]; }
        out[0] = (float)fabs(m1 - m2);
    }
}

extern "C" void kernel_launch(void* const* d_in, const int* in_sizes, int n_in,
                              void* d_out, int out_size, void* d_ws, size_t ws_size,
                              hipStream_t stream) {
    const float* img1 = (const float*)d_in[0];
    const float* img2 = (const float*)d_in[1];
    float* partials   = (float*)d_ws;           // 1024 blocks * 2 floats = 8 KB
    (void)in_sizes; (void)n_in; (void)out_size; (void)ws_size;

    moment_partials_kernel<<<2 * BLOCKS_PER_IMG, THREADS, 0, stream>>>(img1, img2, partials);
    moment_final_kernel<<<1, 32, 0, stream>>>(partials, (float*)d_out);
}